// HiddenFermion_19086834663849
// MI455X (gfx1250) — compile-verified
//
#include <hip/hip_runtime.h>

typedef __attribute__((ext_vector_type(2))) float v2f;
typedef __attribute__((ext_vector_type(8))) float v8f;

#define NS 4096
#define NSITES 100
#define NORB 116          // 100 + 16
#define MDIM (NORB*NORB)  // 13456
#define FEAT 512
#define NOUT 1856         // 16*116
#define LDW 117           // padded LDS row stride (gcd(117,64)=1)
#define CK 16             // K-chunk staged in LDS per iteration

__device__ __forceinline__ float selu_f(float v) {
    const float scale = 1.0507009873554805f;
    const float alpha = 1.6732632423543772f;
    return scale * (v > 0.f ? v : alpha * (expf(v) - 1.f));
}

// ---------------------------------------------------------------------------
// Kernel 1: occupied-mode index list + orbital gather into rows 0..99 of M
// ---------------------------------------------------------------------------
__global__ __launch_bounds__(128) void build_orbs_kernel(
    const float* __restrict__ x, const float* __restrict__ mf,
    const float* __restrict__ hf, float* __restrict__ mmat)
{
    __shared__ int idxs[NSITES];
    const int s = blockIdx.x;
    const int lane = threadIdx.x & 31;
    const int wave = threadIdx.x >> 5;

    if (wave == 0) {
        int count = 0;
        for (int base = 0; base < 2 * NSITES; base += 32) {
            int mode = base + lane;
            bool occ = false;
            if (mode < 2 * NSITES) {
                float xv = x[(size_t)s * NSITES + (mode < NSITES ? mode : mode - NSITES)];
                occ = (mode < NSITES) ? (xv == 1.0f) : (xv == -1.0f);
            }
            unsigned mask = (unsigned)(__ballot(occ) & 0xffffffffull);
            int pre = __popc(mask & ((1u << lane) - 1u));
            if (occ) idxs[count + pre] = mode;
            count += __popc(mask);
        }
    }
    __syncthreads();

    float* out = mmat + (size_t)s * MDIM;
    for (int e = threadIdx.x; e < NSITES * NORB; e += blockDim.x) {
        int r = e / NORB, c = e - r * NORB;
        int i = idxs[r];
        float v = (c < NSITES) ? mf[(size_t)i * NSITES + c]
                               : hf[(size_t)i * 16 + (c - NSITES)];
        out[r * NORB + c] = v;
    }
}

// ---------------------------------------------------------------------------
// Kernel 2: FP32 WMMA GEMM with LDS-staged (double-buffered) B tile held in
// K-pair-interleaved layout so each lane's {B[k][n], B[k+1][n]} is one
// contiguous ds_load_b64 landing in the adjacent VGPR pair WMMA consumes.
// Block = 256 threads = 8 waves; block tile 128(M) x 64(N); each wave owns a
// 16x64 slice via V_WMMA_F32_16X16X4_F32.
// MODE 0: C = selu(A@B).  MODE 1: A@B + bias + eye scattered into rows
// 100..115 of the per-sample 116x116 matrix buffer.
// ---------------------------------------------------------------------------
template <int MODE>
__global__ __launch_bounds__(256) void gemm_wmma_kernel(
    const float* __restrict__ A, const float* __restrict__ B,
    float* __restrict__ C, const float* __restrict__ bias,
    int M, int K, int N)
{
    __shared__ float Bs[2][CK * 64];   // two 4 KB buffers, pair-interleaved

    const int tid  = threadIdx.x;
    const int lane = tid & 31;
    const int wave = tid >> 5;
    const int lh   = lane >> 4;        // half-wave selector -> K base 2*lh
    const int lr   = lane & 15;

    const int m_base = blockIdx.x * 128 + wave * 16;
    const int n_base = blockIdx.y * 64;

    v8f acc[4] = {};
    const float* Arow = A + (size_t)(m_base + lr) * K + 2 * lh;

    const int nchunks = (K + CK - 1) / CK;

    // stage one 16x64 B chunk into LDS, interleaved as
    // Bs[(k>>1)*128 + n*2 + (k&1)]  (zero-pad past K)
    auto stage = [&](int buf, int kk) {
        #pragma unroll
        for (int e = tid; e < CK * 64; e += 256) {
            int kr = e >> 6, nc = e & 63;
            float v = (kk + kr < K)
                    ? B[(size_t)(kk + kr) * N + n_base + nc] : 0.0f;
            Bs[buf][(kr >> 1) * 128 + nc * 2 + (kr & 1)] = v;
        }
    };

    stage(0, 0);
    __syncthreads();

    for (int chunk = 0; chunk < nchunks; ++chunk) {
        const int buf = chunk & 1;
        const int kk  = chunk * CK;
        if (chunk + 1 < nchunks) stage(buf ^ 1, kk + CK);
        __builtin_prefetch(Arow + kk + 2 * CK, 0, 0);   // global_prefetch_b8

        #pragma unroll
        for (int j = 0; j < CK / 4; ++j) {
            int ka = kk + 4 * j;
            int kas = (ka + 4 <= K) ? ka : (K - 4);     // clamp: B is zero there
            v2f a = *(const v2f*)(Arow + kas);          // A[m][k0], A[m][k0+1]
            // kpair = 2j + lh; contiguous {B[k0][n], B[k0+1][n]} per lane
            const float* bl = &Bs[buf][(2 * j + lh) * 128 + lr * 2];
            v2f b0 = *(const v2f*)(bl + 0);
            v2f b1 = *(const v2f*)(bl + 32);
            v2f b2 = *(const v2f*)(bl + 64);
            v2f b3 = *(const v2f*)(bl + 96);
            acc[0] = __builtin_amdgcn_wmma_f32_16x16x4_f32(false, a, false, b0, (short)0, acc[0], false, false);
            acc[1] = __builtin_amdgcn_wmma_f32_16x16x4_f32(false, a, false, b1, (short)0, acc[1], false, false);
            acc[2] = __builtin_amdgcn_wmma_f32_16x16x4_f32(false, a, false, b2, (short)0, acc[2], false, false);
            acc[3] = __builtin_amdgcn_wmma_f32_16x16x4_f32(false, a, false, b3, (short)0, acc[3], false, false);
        }
        __syncthreads();   // staging of buf^1 done; reads of buf done
    }

    #pragma unroll
    for (int t = 0; t < 4; ++t) {
        #pragma unroll
        for (int i = 0; i < 8; ++i) {
            int row = m_base + i + 8 * lh;   // sample index
            int col = n_base + t * 16 + lr;
            float v = acc[t][i];
            if (MODE == 0) {
                C[(size_t)row * N + col] = selu_f(v);
            } else {
                v += bias[col];
                int hid = col / NORB;
                int orb = col - hid * NORB;
                if (orb == NSITES + hid) v += 1.0f;   // identity block
                C[(size_t)row * MDIM + (size_t)(NSITES + hid) * NORB + orb] = v;
            }
        }
    }
}

// ---------------------------------------------------------------------------
// Kernel 3: partial-pivot LU in LDS -> log|det| (re) and pi*[sign<0] (im)
// ---------------------------------------------------------------------------
__global__ __launch_bounds__(128) void lu_det_kernel(
    const float* __restrict__ mmat, float* __restrict__ out)
{
    __shared__ float Am[NORB * LDW];
    __shared__ float red_val[128];
    __shared__ int   red_idx[128];
    __shared__ int   s_piv;

    const int s   = blockIdx.x;
    const int tid = threadIdx.x;
    const float* src = mmat + (size_t)s * MDIM;

    for (int e = tid; e < MDIM; e += 128) {
        int r = e / NORB, c = e - r * NORB;
        Am[r * LDW + c] = src[e];
    }
    __syncthreads();

    float logdet = 0.f;
    int   neg = 0;
    bool  zero = false;

    for (int k = 0; k < NORB; ++k) {
        // --- pivot search: argmax |A[r][k]|, r in [k, 116) ---
        float best = -1.f; int bidx = k;
        for (int r = k + tid; r < NORB; r += 128) {
            float v = fabsf(Am[r * LDW + k]);
            if (v > best) { best = v; bidx = r; }
        }
        red_val[tid] = best; red_idx[tid] = bidx;
        __syncthreads();
        for (int off = 64; off > 0; off >>= 1) {
            if (tid < off) {
                if (red_val[tid + off] > red_val[tid]) {
                    red_val[tid] = red_val[tid + off];
                    red_idx[tid] = red_idx[tid + off];
                }
            }
            __syncthreads();
        }
        if (tid == 0) s_piv = red_idx[0];
        __syncthreads();

        const int p = s_piv;
        if (p != k) {
            for (int c = tid; c < NORB; c += 128) {
                float tmp = Am[k * LDW + c];
                Am[k * LDW + c] = Am[p * LDW + c];
                Am[p * LDW + c] = tmp;
            }
        }
        __syncthreads();

        float piv = Am[k * LDW + k];
        if (p != k)    neg ^= 1;
        if (piv < 0.f) neg ^= 1;
        if (piv == 0.f) zero = true;
        logdet += logf(fabsf(piv));
        float invp = 1.0f / piv;

        // --- rank-1 update of trailing submatrix ---
        for (int r = k + 1 + tid; r < NORB; r += 128) {
            float f = Am[r * LDW + k] * invp;
            Am[r * LDW + k] = f;
            for (int c = k + 1; c < NORB; ++c)
                Am[r * LDW + c] -= f * Am[k * LDW + c];
        }
        __syncthreads();
    }

    if (tid == 0) {
        float re = logdet;
        float im = (neg & 1) ? 3.14159265358979323846f : 0.0f;
        if (zero) { re = -__builtin_inff(); im = 0.0f; }
        out[2 * s + 0] = re;
        out[2 * s + 1] = im;
    }
}

// ---------------------------------------------------------------------------
extern "C" void kernel_launch(void* const* d_in, const int* in_sizes, int n_in,
                              void* d_out, int out_size, void* d_ws, size_t ws_size,
                              hipStream_t stream) {
    const float* x     = (const float*)d_in[0];
    const float* mf    = (const float*)d_in[1];
    const float* hf    = (const float*)d_in[2];
    const float* w0    = (const float*)d_in[3];
    const float* w1    = (const float*)d_in[4];
    const float* w_out = (const float*)d_in[5];
    const float* b_out = (const float*)d_in[6];
    float* out = (float*)d_out;

    float* ws   = (float*)d_ws;
    float* mmat = ws;                                   // 4096*13456 floats
    float* h1   = mmat + (size_t)NS * MDIM;             // 4096*512
    float* h2   = h1 + (size_t)NS * FEAT;               // 4096*512

    // rows 0..99 of each per-sample matrix: gathered orbitals
    build_orbs_kernel<<<NS, 128, 0, stream>>>(x, mf, hf, mmat);

    // h1 = selu(x @ w0)           (4096 x 100) @ (100 x 512)
    gemm_wmma_kernel<0><<<dim3(NS / 128, FEAT / 64), 256, 0, stream>>>(
        x, w0, h1, nullptr, NS, NSITES, FEAT);

    // h2 = selu(h1 @ w1)          (4096 x 512) @ (512 x 512)
    gemm_wmma_kernel<0><<<dim3(NS / 128, FEAT / 64), 256, 0, stream>>>(
        h1, w1, h2, nullptr, NS, FEAT, FEAT);

    // rows 100..115: h2 @ w_out + b_out (+eye), scattered
    gemm_wmma_kernel<1><<<dim3(NS / 128, NOUT / 64), 256, 0, stream>>>(
        h2, w_out, mmat, b_out, NS, FEAT, NOUT);

    // slogdet per sample
    lu_det_kernel<<<NS, 128, 0, stream>>>(mmat, out);
}